// TFKNeighborsClassifier_49057116455120
// MI455X (gfx1250) — compile-verified
//
#include <hip/hip_runtime.h>
#include <cfloat>

typedef float v2f __attribute__((ext_vector_type(2)));
typedef float v8f __attribute__((ext_vector_type(8)));

#define DIMS 128
#define KSEL 5
#define NB2  256   // blocks in top5 pass

// ---------------------------------------------------------------------------
// Phase 1: squared L2 distance of 16 rows per wave via V_WMMA_F32_16X16X4_F32
// using the A==B self-transpose trick: C = A*A^T, diag(C) = row norms of A.
// ---------------------------------------------------------------------------
__device__ __forceinline__ float v8_get(v8f v, int i) {
    float r = v[0];
#pragma unroll
    for (int j = 1; j < 8; ++j) r = (i == j) ? v[j] : r;
    return r;
}

__global__ __launch_bounds__(256)
void knn_dist_wmma(const float* __restrict__ x, const float* __restrict__ q,
                   float* __restrict__ dist, int n)
{
    __shared__ float qs[DIMS];
    const int tid = threadIdx.x;
    if (tid < DIMS) qs[tid] = q[tid];
    __syncthreads();

    const int lane = tid & 31;
    const int wave = tid >> 5;
    const int m    = lane & 15;   // row within tile
    const int h    = lane >> 4;   // K half: columns 2h, 2h+1

    const int  tile    = blockIdx.x * 8 + wave;           // 16 rows per tile
    const long rowBase = (long)tile * 16;
    if (rowBase >= n) return;                             // wave-uniform exit

    long row = rowBase + m;
    if (row >= n) row = n - 1;                            // clamp pad rows
    const float* xr = x + row * DIMS + 2 * h;

    v8f acc = {};
#pragma unroll 8
    for (int d = 0; d < DIMS; d += 4) {
        v2f xv = *(const v2f*)(xr + d);
        v2f qv = *(const v2f*)(&qs[d + 2 * h]);
        v2f a;
        a.x = xv.x - qv.x;
        a.y = xv.y - qv.y;
        // D = A * A^T + C ; diag accumulates sum((x-q)^2) over this K chunk
        acc = __builtin_amdgcn_wmma_f32_16x16x4_f32(
                  false, a, false, a, (short)0, acc, false, false);
    }

    // diag i<8 : vgpr i,   lane i
    // diag i>=8: vgpr i-8, lane i+16
    int outIdx = -1, vsel = 0;
    if (lane < 8)        { outIdx = lane;      vsel = lane;      }
    else if (lane >= 24) { outIdx = lane - 16; vsel = lane - 24; }
    if (outIdx >= 0) {
        long r = rowBase + outIdx;
        if (r < n) dist[r] = sqrtf(v8_get(acc, vsel));
    }
}

// ---------------------------------------------------------------------------
// Top-5 machinery: sorted-ascending (dist, idx) lists, lexicographic compare
// matching jax.lax.top_k tie-breaking (lower index wins ties).
// ---------------------------------------------------------------------------
__device__ __forceinline__ bool cand_less(float d1, int i1, float d2, int i2) {
    return (d1 < d2) || (d1 == d2 && i1 < i2);
}

__device__ __forceinline__ void insert5(float d, int i, float* bd, int* bi) {
    if (!cand_less(d, i, bd[4], bi[4])) return;
    bd[4] = d; bi[4] = i;
#pragma unroll
    for (int j = 4; j > 0; --j) {
        if (cand_less(bd[j], bi[j], bd[j - 1], bi[j - 1])) {
            float td = bd[j]; bd[j] = bd[j - 1]; bd[j - 1] = td;
            int   ti = bi[j]; bi[j] = bi[j - 1]; bi[j - 1] = ti;
        }
    }
}

// log2 tree reduction of per-thread top5 lists through LDS; result in
// thread 0's registers.
__device__ __forceinline__ void block_reduce_top5(float* bd, int* bi,
                                                  float* sd, int* si)
{
    const int tid = threadIdx.x;
#pragma unroll
    for (int j = 0; j < KSEL; ++j) { sd[tid * KSEL + j] = bd[j]; si[tid * KSEL + j] = bi[j]; }
    __syncthreads();
    for (int s = 128; s > 0; s >>= 1) {
        if (tid < s) {
#pragma unroll
            for (int j = 0; j < KSEL; ++j)
                insert5(sd[(tid + s) * KSEL + j], si[(tid + s) * KSEL + j], bd, bi);
#pragma unroll
            for (int j = 0; j < KSEL; ++j) { sd[tid * KSEL + j] = bd[j]; si[tid * KSEL + j] = bi[j]; }
        }
        __syncthreads();
    }
}

// Phase 2: per-block top5 over the distance array.
__global__ __launch_bounds__(256)
void knn_top5_blocks(const float* __restrict__ dist, int n,
                     float* __restrict__ candD, int* __restrict__ candI)
{
    __shared__ float sd[256 * KSEL];
    __shared__ int   si[256 * KSEL];
    float bd[KSEL]; int bi[KSEL];
#pragma unroll
    for (int j = 0; j < KSEL; ++j) { bd[j] = FLT_MAX; bi[j] = 0x7FFFFFFF; }

    for (int i = blockIdx.x * blockDim.x + threadIdx.x; i < n;
         i += gridDim.x * blockDim.x)
        insert5(dist[i], i, bd, bi);

    block_reduce_top5(bd, bi, sd, si);

    if (threadIdx.x == 0) {
#pragma unroll
        for (int j = 0; j < KSEL; ++j) {
            candD[blockIdx.x * KSEL + j] = bd[j];
            candI[blockIdx.x * KSEL + j] = bi[j];
        }
    }
}

// Phase 3: merge candidates, weighted-label voting (faithful broadcast:
// weighted[i][j] = y[idx_i][j] / (-dist_j)), argmax, one-hot output.
__global__ __launch_bounds__(256)
void knn_finalize(const float* __restrict__ candD, const int* __restrict__ candI,
                  int nc, const float* __restrict__ y, float* __restrict__ out)
{
    __shared__ float sd[256 * KSEL];
    __shared__ int   si[256 * KSEL];
    __shared__ int   result;
    float bd[KSEL]; int bi[KSEL];
#pragma unroll
    for (int j = 0; j < KSEL; ++j) { bd[j] = FLT_MAX; bi[j] = 0x7FFFFFFF; }

    for (int i = threadIdx.x; i < nc; i += blockDim.x)
        insert5(candD[i], candI[i], bd, bi);

    block_reduce_top5(bd, bi, sd, si);

    if (threadIdx.x == 0) {
        // bd ascending distance == neg_dists descending (top_k order)
        float best = -FLT_MAX; int bestI = 0;
        for (int i = 0; i < KSEL; ++i) {
            float s = 0.0f;
            for (int j = 0; j < KSEL; ++j)
                s += y[(long)bi[i] * KSEL + j] / (-bd[j]);
            if (s > best) { best = s; bestI = i; }   // strict > : first max
        }
        result = bestI;
    }
    __syncthreads();
    if (threadIdx.x < KSEL)
        out[threadIdx.x] = (threadIdx.x == result) ? 1.0f : 0.0f;
}

// ---------------------------------------------------------------------------
extern "C" void kernel_launch(void* const* d_in, const int* in_sizes, int n_in,
                              void* d_out, int out_size, void* d_ws, size_t ws_size,
                              hipStream_t stream)
{
    const float* x = (const float*)d_in[0];   // [N, 128]
    const float* y = (const float*)d_in[1];   // [N, 5]
    const float* q = (const float*)d_in[2];   // [128]
    // d_in[3] = k (always 5 per reference)

    const int n = in_sizes[0] / DIMS;

    float* dist  = (float*)d_ws;                                  // N floats
    float* candD = (float*)((char*)d_ws + (size_t)n * sizeof(float));
    int*   candI = (int*)(candD + NB2 * KSEL);

    const int tiles   = (n + 15) / 16;
    const int blocks1 = (tiles + 7) / 8;   // 8 waves (tiles) per 256-thread block

    knn_dist_wmma <<<blocks1, 256, 0, stream>>>(x, q, dist, n);
    knn_top5_blocks<<<NB2,     256, 0, stream>>>(dist, n, candD, candI);
    knn_finalize  <<<1,        256, 0, stream>>>(candD, candI, NB2 * KSEL, y,
                                                 (float*)d_out);
}